// PostProcessor_25074019074087
// MI455X (gfx1250) — compile-verified
//
#include <hip/hip_runtime.h>
#include <stdint.h>

#define HH    192
#define WW    640
#define HW    (HH * WW)
#define NCLS  3
#define BB    32
#define KDET  100
#define NREG  8
#define ROWS  8
#define NSTRIPS (HH / ROWS)
#define TILE_ROWS (ROWS + 2)
#define TILE_ELE (TILE_ROWS * WW)
#define CHUNKS_PER_ROW (WW / 4)            /* B128 = 4 floats */
#define STRIP_CHUNKS (TILE_ROWS * CHUNKS_PER_ROW)   /* 1600 */
#define SELCAP 1024
#define TPB_A  1024
#define NWAVES (TPB_A / 32)                /* 32 */
#define CHUNKS_PER_WAVE (STRIP_CHUNKS / NWAVES)     /* 50 */
#define TPB_B  512

#define PI_F   3.14159265358979323846f
#define HPI_F  1.57079632679489662f

// ---------------------------------------------------------------------------
// CDNA5 async global->LDS staging (gfx1250): data lands directly in LDS,
// tracked by ASYNCcnt; no VGPR round trip.
// ---------------------------------------------------------------------------
__device__ __forceinline__ void async_load_b128(unsigned lds_byte_addr,
                                                unsigned gl_byte_off,
                                                const void* base) {
  // GVS mode: mem_addr = SADDR(64) + VADDR(32) ; LDS addr from VDST vgpr.
  asm volatile("global_load_async_to_lds_b128 %0, %1, %2"
               :: "v"(lds_byte_addr), "v"(gl_byte_off), "s"(base)
               : "memory");
}
__device__ __forceinline__ void wait_async_le2() {
  asm volatile("s_wait_asynccnt 2" ::: "memory");
}
__device__ __forceinline__ void wait_async_zero() {
  asm volatile("s_wait_asynccnt 0" ::: "memory");
}

// Issue async loads for rows [r0-1 .. r0+ROWS] (edge-clamped) into `tile`.
// Chunk assignment guarantees EVERY wave issues exactly 2 async instructions,
// so "s_wait_asynccnt 2" later implies this strip has fully landed (per-wave
// in-order completion of async loads).
__device__ __forceinline__ void issue_strip(const float* __restrict__ plane,
                                            int r0, float* tile) {
  const int lane = threadIdx.x & 31;
  const int wave = threadIdx.x >> 5;
  const int base = wave * CHUNKS_PER_WAVE;        // 50 chunks per wave
#pragma unroll
  for (int it = 0; it < 2; ++it) {
    int c = base + it * 32 + lane;
    bool act = (it == 0) | (lane < (CHUNKS_PER_WAVE - 32));  // 18 lanes in 2nd
    if (act) {
      int j = c / CHUNKS_PER_ROW;
      int k = c - j * CHUNKS_PER_ROW;
      int gr = r0 - 1 + j;
      gr = gr < 0 ? 0 : (gr > HH - 1 ? HH - 1 : gr);
      unsigned laddr = (unsigned)(uintptr_t)(&tile[j * WW + k * 4]);
      unsigned goff  = (unsigned)((gr * WW + k * 4) * 4);
      async_load_b128(laddr, goff, (const void*)plane);
    }
  }
}

__device__ __forceinline__ float nms_val(const float* tile, int dy, int x) {
  int ty = dy + 1;
  int xm = x > 0 ? x - 1 : 0;
  int xp = x < WW - 1 ? x + 1 : WW - 1;
  const float* r0p = &tile[(ty - 1) * WW];
  const float* r1p = &tile[ty * WW];
  const float* r2p = &tile[(ty + 1) * WW];
  float v = r1p[x];
  float m = fmaxf(fmaxf(r0p[xm], r0p[x]), r0p[xp]);
  m = fmaxf(m, fmaxf(r1p[xm], fmaxf(v, r1p[xp])));
  m = fmaxf(m, fmaxf(fmaxf(r2p[xm], r2p[x]), r2p[xp]));
  return (v == m) ? v : 0.0f;
}

// ---------------------------------------------------------------------------
// Kernel A: per-(b,c) plane NMS + exact top-100.
// Radix-select (4x8-bit MSB-first, zeros excluded) + collect + bitonic sort.
// Double-buffered async-LDS pipeline overlaps the plane stream with the
// 3x3 stencil + histogram compute.
// ---------------------------------------------------------------------------
__global__ __launch_bounds__(TPB_A)
void nms_topk_plane(const float* __restrict__ heat,
                    float* __restrict__ ws_vals, int* __restrict__ ws_idx) {
  __shared__ __align__(16) float tile[2][TILE_ELE];      // 51.2 KB
  __shared__ unsigned hist[256];
  __shared__ unsigned long long sel[SELCAP];             // 8 KB
  __shared__ unsigned sh_pref, sh_rem, sh_cnt, sh_done;

  const int p   = blockIdx.x;                 // plane = b*NCLS + c
  const int tid = threadIdx.x;
  const float* plane = heat + (size_t)p * HW;

  unsigned prefix = 0u, rem = KDET, done = 0u;
  for (int pass = 0; pass < 4 && !done; ++pass) {
    const int shift = 24 - 8 * pass;
    const unsigned hmask = (pass == 0) ? 0u : (0xFFFFFFFFu << (32 - 8 * pass));
    for (int i = tid; i < 256; i += TPB_A) hist[i] = 0u;
    __syncthreads();

    issue_strip(plane, 0, tile[0]);
    for (int s = 0; s < NSTRIPS; ++s) {
      const bool last = (s == NSTRIPS - 1);
      if (!last) issue_strip(plane, (s + 1) * ROWS, tile[(s + 1) & 1]);
      if (!last) wait_async_le2(); else wait_async_zero();
      __syncthreads();                       // strip s visible to all waves
      const float* tl = tile[s & 1];
      for (int e = tid; e < ROWS * WW; e += TPB_A) {
        int dy = e / WW, x = e - dy * WW;
        unsigned bits = __float_as_uint(nms_val(tl, dy, x));
        // skip zeros: they can't matter unless positives < K (handled below)
        if (bits != 0u && (bits & hmask) == prefix)
          atomicAdd(&hist[(bits >> shift) & 0xFFu], 1u);
      }
      __syncthreads();                       // done reading before buf reuse
    }
    if (tid == 0) {
      unsigned acc = 0u, r = rem, pf = prefix, found = 0u;
      for (int d = 255; d >= 0; --d) {
        unsigned c = hist[d];
        if (acc + c >= r) { pf |= ((unsigned)d) << shift; r -= acc; found = 1u; break; }
        acc += c;
      }
      if (!found) {   // fewer than K positive peaks: take all positives (T=1)
        sh_pref = 1u; sh_rem = 0u; sh_done = 1u;
      } else {
        sh_pref = pf; sh_rem = r; sh_done = 0u;
      }
    }
    __syncthreads();
    prefix = sh_pref; rem = sh_rem; done = sh_done;
    __syncthreads();
  }
  const unsigned T = prefix;   // bit pattern of the 100th-largest NMS value

  if (tid == 0) sh_cnt = 0u;
  __syncthreads();
  issue_strip(plane, 0, tile[0]);
  for (int s = 0; s < NSTRIPS; ++s) {
    const bool last = (s == NSTRIPS - 1);
    if (!last) issue_strip(plane, (s + 1) * ROWS, tile[(s + 1) & 1]);
    if (!last) wait_async_le2(); else wait_async_zero();
    __syncthreads();
    const float* tl = tile[s & 1];
    for (int e = tid; e < ROWS * WW; e += TPB_A) {
      int dy = e / WW, x = e - dy * WW;
      unsigned bits = __float_as_uint(nms_val(tl, dy, x));
      if (bits != 0u && bits >= T) {
        unsigned pos = atomicAdd(&sh_cnt, 1u);
        if (pos < SELCAP) {
          unsigned hw = (unsigned)((s * ROWS + dy) * WW + x);
          // key: value descending, then lower index first (jax tie-break)
          sel[pos] = ((unsigned long long)bits << 32) |
                     (unsigned long long)(0xFFFFFFFFu - hw);
        }
      }
    }
    __syncthreads();
  }
  unsigned cnt = sh_cnt < SELCAP ? sh_cnt : SELCAP;
  for (int i = tid; i < SELCAP; i += TPB_A)
    if ((unsigned)i >= cnt) sel[i] = 0ull;
  __syncthreads();

  // bitonic sort, descending (1024 threads, 1024 keys)
  for (int k2 = 2; k2 <= SELCAP; k2 <<= 1)
    for (int j = k2 >> 1; j > 0; j >>= 1) {
      int i = tid, ixj = i ^ j;
      if (ixj > i) {
        unsigned long long a = sel[i], b = sel[ixj];
        bool desc = ((i & k2) == 0);
        if (desc ? (a < b) : (a > b)) { sel[i] = b; sel[ixj] = a; }
      }
      __syncthreads();
    }

  if (tid < KDET) {
    unsigned long long key = sel[tid];
    unsigned bits = (unsigned)(key >> 32);
    unsigned hw = bits ? (0xFFFFFFFFu - (unsigned)(key & 0xFFFFFFFFull)) : 0u;
    ws_vals[p * KDET + tid] = __uint_as_float(bits);
    ws_idx [p * KDET + tid] = (int)hw;
  }
}

// ---------------------------------------------------------------------------
// Kernel B: per-batch merge of 3x100 -> top-100 (stable by flat index c*K+r)
// ---------------------------------------------------------------------------
__global__ __launch_bounds__(TPB_B)
void merge_batch(const float* __restrict__ ws_vals, const int* __restrict__ ws_idx,
                 float* __restrict__ mscore, int* __restrict__ mcls,
                 int* __restrict__ mhw) {
  __shared__ unsigned long long key[TPB_B];
  const int b = blockIdx.x, tid = threadIdx.x;
  unsigned long long k = 0ull;
  if (tid < NCLS * KDET) {
    unsigned bits = __float_as_uint(ws_vals[b * NCLS * KDET + tid]);
    if (bits)
      k = ((unsigned long long)bits << 32) | (unsigned long long)(1023u - (unsigned)tid);
  }
  key[tid] = k;
  __syncthreads();
  for (int k2 = 2; k2 <= TPB_B; k2 <<= 1)
    for (int j = k2 >> 1; j > 0; j >>= 1) {
      int i = tid, ixj = i ^ j;
      if (ixj > i) {
        unsigned long long a = key[i], c = key[ixj];
        bool desc = ((i & k2) == 0);
        if (desc ? (a < c) : (a > c)) { key[i] = c; key[ixj] = a; }
      }
      __syncthreads();
    }
  if (tid < KDET) {
    unsigned long long kk = key[tid];
    unsigned bits = (unsigned)(kk >> 32);
    int o = b * KDET + tid;
    if (bits) {
      unsigned slot = 1023u - (unsigned)(kk & 0xFFFFFFFFull); // = c*KDET + rank
      mscore[o] = __uint_as_float(bits);
      mcls  [o] = (int)(slot / KDET);
      mhw   [o] = ws_idx[b * NCLS * KDET + slot];
    } else {
      mscore[o] = 0.0f; mcls[o] = 0; mhw[o] = 0;  // score<=thr rows zeroed anyway
    }
  }
}

// ---------------------------------------------------------------------------
// Kernel C: per-detection decode (gather regs, inverses, 3D box, 2D box)
// ---------------------------------------------------------------------------
__device__ __forceinline__ void inv3(const float* a, float* o) {
  float det = a[0] * (a[4] * a[8] - a[5] * a[7])
            - a[1] * (a[3] * a[8] - a[5] * a[6])
            + a[2] * (a[3] * a[7] - a[4] * a[6]);
  float id = 1.0f / det;
  o[0] = (a[4] * a[8] - a[5] * a[7]) * id;
  o[1] = (a[2] * a[7] - a[1] * a[8]) * id;
  o[2] = (a[1] * a[5] - a[2] * a[4]) * id;
  o[3] = (a[5] * a[6] - a[3] * a[8]) * id;
  o[4] = (a[0] * a[8] - a[2] * a[6]) * id;
  o[5] = (a[2] * a[3] - a[0] * a[5]) * id;
  o[6] = (a[3] * a[7] - a[4] * a[6]) * id;
  o[7] = (a[1] * a[6] - a[0] * a[7]) * id;
  o[8] = (a[0] * a[4] - a[1] * a[3]) * id;
}

__global__ void decode_dets(const float* __restrict__ reg,
                            const float* __restrict__ Kmat,
                            const float* __restrict__ Tmat,
                            const float* __restrict__ size2,
                            const float* __restrict__ mscore,
                            const int* __restrict__ mcls,
                            const int* __restrict__ mhw,
                            float* __restrict__ out) {
  int det = blockIdx.x * blockDim.x + threadIdx.x;
  if (det >= BB * KDET) return;
  const int b = det / KDET;

  float score = mscore[det];
  int   cls   = mcls[det];
  int   hw    = mhw[det];
  float xs = (float)(hw % WW), ys = (float)(hw / WW);

  const float* rp = reg + (size_t)b * NREG * HW + hw;
  float r[NREG];
#pragma unroll
  for (int ch = 0; ch < NREG; ++ch) r[ch] = rp[(size_t)ch * HW];

  float depth = r[0] * 16.32f + 28.01f;

  float Tm[9], Km[9], Ti[9], Ki[9];
#pragma unroll
  for (int i = 0; i < 9; ++i) { Tm[i] = Tmat[b * 9 + i]; Km[i] = Kmat[b * 9 + i]; }
  inv3(Tm, Ti);
  inv3(Km, Ki);

  float ppx = xs + r[1], ppy = ys + r[2];
  float ix = (Ti[0] * ppx + Ti[1] * ppy + Ti[2]) * depth;
  float iy = (Ti[3] * ppx + Ti[4] * ppy + Ti[5]) * depth;
  float iz = (Ti[6] * ppx + Ti[7] * ppy + Ti[8]) * depth;
  float lx = Ki[0] * ix + Ki[1] * iy + Ki[2] * iz;
  float ly = Ki[3] * ix + Ki[4] * iy + Ki[5] * iz;
  float lz = Ki[6] * ix + Ki[7] * iy + Ki[8] * iz;

  const float DIMREF[3][3] = {{3.88f, 1.63f, 1.53f},
                              {1.78f, 1.70f, 0.58f},
                              {0.88f, 1.73f, 0.67f}};
  int ci = cls < 0 ? 0 : (cls > 2 ? 2 : cls);
  float d0 = __expf(r[3]) * DIMREF[ci][0];
  float d1 = __expf(r[4]) * DIMREF[ci][1];
  float d2 = __expf(r[5]) * DIMREF[ci][2];
  ly += 0.5f * d1;

  float ray   = atanf(lx / (lz + 1e-7f));
  float alpha = atanf(r[6] / (r[7] + 1e-7f)) + (r[7] >= 0.0f ? -HPI_F : HPI_F);
  float roty  = alpha + ray;
  if (roty >  PI_F) roty -= 2.0f * PI_F;
  if (roty < -PI_F) roty += 2.0f * PI_F;

  // corner coefficients = BOX_SIGNS[i][BOX_IDX[i][k]] (precomputed)
  const float C0[8] = {-0.5f, 0.5f, 0.5f, 0.5f, 0.5f, -0.5f, -0.5f, -0.5f};
  const float C1[8] = {-1.f, -1.f, 0.f, 0.f, -1.f, -1.f, 0.f, 0.f};
  const float C2[8] = {-0.5f, -0.5f, -0.5f, 0.5f, 0.5f, 0.5f, 0.5f, -0.5f};
  float cr = cosf(roty), sr = sinf(roty);
  float umin = 1e30f, umax = -1e30f, vmin = 1e30f, vmax = -1e30f;
#pragma unroll
  for (int k = 0; k < 8; ++k) {
    float bx = d0 * C0[k], by = d1 * C1[k], bz = d2 * C2[k];
    float wx =  cr * bx + sr * bz + lx;
    float wy =  by + ly;
    float wz = -sr * bx + cr * bz + lz;
    float px = Km[0] * wx + Km[1] * wy + Km[2] * wz;
    float py = Km[3] * wx + Km[4] * wy + Km[5] * wz;
    float pz = Km[6] * wx + Km[7] * wy + Km[8] * wz;
    float u = px / pz, v = py / pz;
    umin = fminf(umin, u); umax = fmaxf(umax, u);
    vmin = fminf(vmin, v); vmax = fmaxf(vmax, v);
  }
  float szx = size2[0], szy = size2[1];   // reference uses sz[0], sz[1] globally
  float xmin = fminf(fmaxf(umin, 0.0f), szx);
  float xmax = fminf(fmaxf(umax, 0.0f), szx);
  float ymin = fminf(fmaxf(vmin, 0.0f), szy);
  float ymax = fminf(fmaxf(vmax, 0.0f), szy);

  float keep = (score > 0.25f) ? 1.0f : 0.0f;
  float* o = out + (size_t)det * 14;
  o[0]  = (float)cls * keep;
  o[1]  = alpha * keep;
  o[2]  = xmin * keep;
  o[3]  = ymin * keep;
  o[4]  = xmax * keep;
  o[5]  = ymax * keep;
  o[6]  = d1 * keep;     // dims rolled by -1: [d1, d2, d0]
  o[7]  = d2 * keep;
  o[8]  = d0 * keep;
  o[9]  = lx * keep;
  o[10] = ly * keep;
  o[11] = lz * keep;
  o[12] = roty * keep;
  o[13] = score * keep;
}

// ---------------------------------------------------------------------------
extern "C" void kernel_launch(void* const* d_in, const int* in_sizes, int n_in,
                              void* d_out, int out_size, void* d_ws, size_t ws_size,
                              hipStream_t stream) {
  const float* heat  = (const float*)d_in[0];   // (32,3,192,640)
  const float* reg   = (const float*)d_in[1];   // (32,8,192,640)
  const float* Kmat  = (const float*)d_in[2];   // (32,3,3)
  const float* Tmat  = (const float*)d_in[3];   // (32,3,3)
  const float* size2 = (const float*)d_in[4];   // (32,2)

  float* ws_vals = (float*)d_ws;                              // 96*100
  int*   ws_idx  = (int*)(ws_vals + BB * NCLS * KDET);        // 96*100
  float* mscore  = (float*)(ws_idx + BB * NCLS * KDET);       // 32*100
  int*   mcls    = (int*)(mscore + BB * KDET);                // 32*100
  int*   mhw     = (int*)(mcls + BB * KDET);                  // 32*100

  nms_topk_plane<<<BB * NCLS, TPB_A, 0, stream>>>(heat, ws_vals, ws_idx);
  merge_batch<<<BB, TPB_B, 0, stream>>>(ws_vals, ws_idx, mscore, mcls, mhw);
  decode_dets<<<(BB * KDET + 255) / 256, 256, 0, stream>>>(
      reg, Kmat, Tmat, size2, mscore, mcls, mhw, (float*)d_out);
}